// Boyd_MPO_59081570125223
// MI455X (gfx1250) — compile-verified
//
#include <hip/hip_runtime.h>
#include <math.h>

typedef float v2f __attribute__((ext_vector_type(2)));
typedef float v8f __attribute__((ext_vector_type(8)));

#define NB      128      // batch
#define Nn      500      // assets
#define NPAD    512      // padded rows (32 tiles of 16)
#define Tt      12       // time steps
#define Tp      16       // padded cols (one WMMA N-tile)
#define ITERS   400
#define LR0     0.05f
#define NTHREADS 512
#define NWAVES   16      // 512 / 32 (wave32)
#define NBISECT  26

__device__ __forceinline__ float wave_sum(float v) {
    v += __shfl_xor(v, 16, 32);
    v += __shfl_xor(v, 8, 32);
    v += __shfl_xor(v, 4, 32);
    v += __shfl_xor(v, 2, 32);
    v += __shfl_xor(v, 1, 32);
    return v;
}
__device__ __forceinline__ float wave_min(float v) {
    v = fminf(v, __shfl_xor(v, 16, 32));
    v = fminf(v, __shfl_xor(v, 8, 32));
    v = fminf(v, __shfl_xor(v, 4, 32));
    v = fminf(v, __shfl_xor(v, 2, 32));
    v = fminf(v, __shfl_xor(v, 1, 32));
    return v;
}
__device__ __forceinline__ float wave_max(float v) {
    v = fmaxf(v, __shfl_xor(v, 16, 32));
    v = fmaxf(v, __shfl_xor(v, 8, 32));
    v = fmaxf(v, __shfl_xor(v, 4, 32));
    v = fmaxf(v, __shfl_xor(v, 2, 32));
    v = fmaxf(v, __shfl_xor(v, 1, 32));
    return v;
}

// One persistent workgroup per batch element (one per WGP); all 400 iterations
// run in-kernel with W/Y/G LDS-resident (96 KB of the 320 KB WGP LDS).
// covmat (1 MB/batch, 128 MB total) stays resident in MI455X's 192 MB L2.
// 512 threads = 16 wave32 waves = 4 waves/SIMD32 to hide VMEM + XDL latency.
// Padded rows 500..511 of Y/G hold garbage (clamped-address loads) but are
// provably never read: both matmul K-loops, the subgradient step, projection
// and the output loop all stop at row 500; padded W rows stay 0 from init.
__global__ __launch_bounds__(NTHREADS)
void boyd_mpo_kernel(const float* __restrict__ rets,
                     const float* __restrict__ covmat,
                     const float* __restrict__ prev_w,
                     float* __restrict__ out) {
    __shared__ float Wl[NPAD * Tp];   // current iterate W (padded)
    __shared__ float Yl[NPAD * Tp];   // Y = C^T W
    __shared__ float Gl[NPAD * Tp];   // M = C Y  (g_quad = 2*M)

    const int b    = blockIdx.x;
    const int tid  = threadIdx.x;
    const int wave = tid >> 5;
    const int lane = tid & 31;
    const int half = lane >> 4;        // 0: lanes 0-15, 1: lanes 16-31
    const int l15  = lane & 15;
    const int koff = half * 2;         // K offset within a 4-chunk (f32 WMMA layout)

    const float* C  = covmat + (size_t)b * Nn * Nn;
    const float* R  = rets   + (size_t)b * Nn * Tt;
    const float* PW = prev_w + (size_t)b * Nn;

    // Output rows for this wave: tiles {wave, wave+16} -> rows m0, m0+256.
    // Clamp to stay in-bounds (garbage rows never read downstream).
    const int m0  = wave * 16 + l15;                  // < 256 always
    const int mc0 = m0;
    const int mc1 = (m0 + 256 < Nn) ? (m0 + 256) : (Nn - 1);

    // ---- init W0 = 1/n on valid (row<500, col<12), 0 on padding ----
    for (int idx = tid; idx < NPAD * Tp; idx += NTHREADS) {
        int row = idx / Tp, col = idx % Tp;
        Wl[idx] = (row < Nn && col < Tt) ? (1.0f / (float)Nn) : 0.0f;
        Yl[idx] = 0.0f;
        Gl[idx] = 0.0f;
    }
    __syncthreads();

    for (int it = 0; it < ITERS; ++it) {
        // ================= pass 1: Y = C^T * W =================
        // Y[j,t] = sum_i C[i*Nn + j] * W[i,t]; A-frag rows come from C columns
        // (coalesced across lanes). One shared B-frag feeds 2 independent WMMAs.
        {
            v8f acc0 = {}, acc1 = {};
            int aoff = koff * Nn;          // element offset of A K-row (k0+koff)
            int wofs = koff * Tp + l15;    // LDS offset of B K-row
            for (int k0 = 0; k0 < Nn; k0 += 4) {     // 125 exact chunks
                v2f bf;
                bf.x = Wl[wofs];
                bf.y = Wl[wofs + Tp];
                v2f a0, a1;
                a0.x = C[aoff + mc0];      a0.y = C[aoff + Nn + mc0];
                a1.x = C[aoff + mc1];      a1.y = C[aoff + Nn + mc1];
                acc0 = __builtin_amdgcn_wmma_f32_16x16x4_f32(false, a0, false, bf, (short)0, acc0, false, false);
                acc1 = __builtin_amdgcn_wmma_f32_16x16x4_f32(false, a1, false, bf, (short)0, acc1, false, false);
                aoff += 4 * Nn;
                wofs += 4 * Tp;
            }
            const int rb = wave * 16 + half * 8;
#pragma unroll
            for (int r = 0; r < 8; ++r) {
                Yl[(rb + r) * Tp + l15]       = acc0[r];
                Yl[(rb + 256 + r) * Tp + l15] = acc1[r];
            }
        }
        __syncthreads();

        // ================= pass 2: M = C * Y =================
        // M[i,t] = sum_j C[i*Nn + j] * Y[j,t]; A-frag = two adjacent floats of
        // row i -> single 8B-aligned b64 load per tile per chunk.
        {
            v8f acc0 = {}, acc1 = {};
            const size_t r0 = (size_t)mc0 * Nn + koff;
            const size_t r1 = (size_t)mc1 * Nn + koff;
            int yofs = koff * Tp + l15;
            for (int k0 = 0; k0 < Nn; k0 += 4) {
                v2f bf;
                bf.x = Yl[yofs];
                bf.y = Yl[yofs + Tp];
                v2f a0 = *(const v2f*)(C + r0 + k0);
                v2f a1 = *(const v2f*)(C + r1 + k0);
                acc0 = __builtin_amdgcn_wmma_f32_16x16x4_f32(false, a0, false, bf, (short)0, acc0, false, false);
                acc1 = __builtin_amdgcn_wmma_f32_16x16x4_f32(false, a1, false, bf, (short)0, acc1, false, false);
                yofs += 4 * Tp;
            }
            const int rb = wave * 16 + half * 8;
#pragma unroll
            for (int r = 0; r < 8; ++r) {
                Gl[(rb + r) * Tp + l15]       = acc0[r];
                Gl[(rb + 256 + r) * Tp + l15] = acc1[r];
            }
        }
        __syncthreads();

        // ================= subgradient step (row per thread) =================
        const float lr = LR0 / sqrtf((float)it + 1.0f);
        for (int rr = tid; rr < Nn; rr += NTHREADS) {
            float w[Tt], s[Tt + 1];
            const float wp = PW[rr];
#pragma unroll
            for (int t = 0; t < Tt; ++t) w[t] = Wl[rr * Tp + t];
            {
                float d = w[0] - wp;
                s[0] = (d > 0.0f) ? 1.0f : ((d < 0.0f) ? -1.0f : 0.0f);
            }
#pragma unroll
            for (int t = 1; t < Tt; ++t) {
                float d = w[t] - w[t - 1];
                s[t] = (d > 0.0f) ? 1.0f : ((d < 0.0f) ? -1.0f : 0.0f);
            }
            s[Tt] = 0.0f;
#pragma unroll
            for (int t = 0; t < Tt; ++t) {
                float g = -R[(size_t)rr * Tt + t]
                          + 2.0f * Gl[rr * Tp + t]
                          + (s[t] - s[t + 1]);
                Wl[rr * Tp + t] = w[t] - lr * g;
            }
        }
        __syncthreads();

        // ========== simplex projection of each time-column (wave per column) ==========
        // Bisection on theta: f(theta) = sum_i max(v_i - theta, 0) - 1 (monotone).
        for (int t = wave; t < Tt; t += NWAVES) {
            float v[16];
            float vmin = 1e30f, vmax = -1e30f;
#pragma unroll
            for (int j = 0; j < 16; ++j) {
                int row = lane + 32 * j;
                bool valid = (row < Nn);
                float x = valid ? Wl[row * Tp + t] : -1e30f;
                v[j] = x;
                vmin = fminf(vmin, valid ? x : 1e30f);
                vmax = fmaxf(vmax, x);
            }
            vmin = wave_min(vmin);
            vmax = wave_max(vmax);
            float lo = vmin - 1.0f / (float)Nn;   // f(lo) >= 0
            float hi = vmax;                      // f(hi) = -1 < 0
            for (int bs = 0; bs < NBISECT; ++bs) {
                float mid = 0.5f * (lo + hi);
                float ps = 0.0f;
#pragma unroll
                for (int j = 0; j < 16; ++j) ps += fmaxf(v[j] - mid, 0.0f);
                float tot = wave_sum(ps);
                if (tot > 1.0f) lo = mid; else hi = mid;
            }
            const float theta = 0.5f * (lo + hi);
#pragma unroll
            for (int j = 0; j < 16; ++j) {
                int row = lane + 32 * j;
                if (row < Nn) Wl[row * Tp + t] = fmaxf(v[j] - theta, 0.0f);
            }
        }
        __syncthreads();
    }

    // ---- write result (B, n, T) row-major ----
    for (int e = tid; e < Nn * Tt; e += NTHREADS) {
        int i = e / Tt, t = e % Tt;
        out[(size_t)b * Nn * Tt + e] = Wl[i * Tp + t];
    }
}

extern "C" void kernel_launch(void* const* d_in, const int* in_sizes, int n_in,
                              void* d_out, int out_size, void* d_ws, size_t ws_size,
                              hipStream_t stream) {
    (void)n_in; (void)d_ws; (void)ws_size; (void)out_size;
    const float* rets   = (const float*)d_in[0];  // (B, n, T) f32
    const float* covmat = (const float*)d_in[1];  // (B, n, n) f32
    const float* prev_w = (const float*)d_in[2];  // (B, n)    f32
    float* out = (float*)d_out;                   // (B, n, T) f32

    int B = in_sizes[2] / Nn;                     // robust batch count (=128)
    hipLaunchKernelGGL(boyd_mpo_kernel, dim3(B), dim3(NTHREADS), 0, stream,
                       rets, covmat, prev_w, out);
}